// SeqAttnContextMLP_20023137534648
// MI455X (gfx1250) — compile-verified
//
#include <hip/hip_runtime.h>
#include <hip/hip_bf16.h>

// Problem constants (match the reference).
#define B_DIM   8
#define LX_DIM  128
#define LY_DIM  512
#define D_DIM   256

typedef __attribute__((ext_vector_type(2))) float v2f;
typedef __attribute__((ext_vector_type(4))) float v4f;
typedef __attribute__((ext_vector_type(8))) float v8f;
typedef __attribute__((ext_vector_type(4))) int   v4i;

// ---------------------------------------------------------------------------
// CDNA5 async global->LDS staging (ASYNCcnt path), with safe fallbacks.
// ---------------------------------------------------------------------------
#if defined(__gfx1250__) && __has_builtin(__builtin_amdgcn_global_load_async_to_lds_b128)
#define HAS_ASYNC_LDS 1
#else
#define HAS_ASYNC_LDS 0
#endif

#if HAS_ASYNC_LDS
#if __has_builtin(__builtin_amdgcn_s_wait_asynccnt)
#define WAIT_ASYNC(n) __builtin_amdgcn_s_wait_asynccnt(n)
#else
#define WAIT_ASYNC(n) asm volatile("s_wait_asynccnt " #n ::: "memory")
#endif
#else
#define WAIT_ASYNC(n)
#endif

// ---------------------------------------------------------------------------
// Kernel 1/2: H = src @ W (+ bias).  f32 WMMA 16x16x4, one 16x16 tile / wave.
// src: M x 256 row-major, W: 256 x 256 row-major, dst: M x 256.
// ---------------------------------------------------------------------------
__global__ __launch_bounds__(128) void h_gemm_kernel(
    const float* __restrict__ src,
    const float* __restrict__ W,
    const float* __restrict__ bias,   // may be nullptr
    float* __restrict__ dst,
    int Mtiles)
{
    const int lane = threadIdx.x & 31;
    const int wave = threadIdx.x >> 5;
    const int tile = blockIdx.x * 4 + wave;
    const int NT   = D_DIM / 16;
    if (tile >= Mtiles * NT) return;
    const int mt = tile / NT;
    const int nt = tile % NT;

    // ISA 7.12.2: A 16x4 f32 -> lane m = lane%16, VGPR pair holds K = bk, bk+1
    // where bk = (lane/16)*2.  B 4x16 -> lane n = lane%16, same K striping.
    const int mn = lane & 15;
    const int bk = (lane >> 4) << 1;

    const float* arow = src + (size_t)(mt * 16 + mn) * D_DIM;
    const float* bcol = W + nt * 16 + mn;

    v8f c = {};
    for (int k = 0; k < D_DIM; k += 4) {
        v2f a = *(const v2f*)(arow + k + bk);           // K=bk, bk+1 (8B aligned)
        v2f b;
        b.x = bcol[(size_t)(k + bk)     * D_DIM];
        b.y = bcol[(size_t)(k + bk + 1) * D_DIM];
        c = __builtin_amdgcn_wmma_f32_16x16x4_f32(
                false, a, false, b, (short)0, c, false, false);
    }

    const float bv = bias ? bias[nt * 16 + mn] : 0.0f;
    const int mbase = mt * 16 + ((lane >> 4) << 3);     // C: VGPR r -> M = r (+8)
    float* drow = dst + nt * 16 + mn;
    #pragma unroll
    for (int r = 0; r < 8; ++r)
        drow[(size_t)(mbase + r) * D_DIM] = c[r] + bv;
}

// ---------------------------------------------------------------------------
// Kernel 3: fused scores -> masked softmax -> contexts, per (b, 16-x-row tile).
// ---------------------------------------------------------------------------
#define CHUNK 64            // y rows staged in LDS at a time
#define HXS   260           // padded row strides (floats) to dodge bank conflicts
#define HYS   260
#define SSS   516

__device__ __forceinline__ float relu_dot4(v4f hx, v4f hy, v4f w) {
    float r = fmaxf(hx.x + hy.x, 0.0f) * w.x;
    r      += fmaxf(hx.y + hy.y, 0.0f) * w.y;
    r      += fmaxf(hx.z + hy.z, 0.0f) * w.z;
    r      += fmaxf(hx.w + hy.w, 0.0f) * w.w;
    return r;
}

// Stage one CHUNK x 256 f32 tile (row stride 256 in global, HYS in LDS).
// Async path: 16 global_load_async_to_lds_b128 per wave -> tracked by ASYNCcnt.
__device__ __forceinline__ void stage_chunk(const float* __restrict__ gsrc,
                                            float* lbase, int tid) {
#if HAS_ASYNC_LDS
    for (int i = tid; i < CHUNK * 64; i += 256) {       // i indexes 16B packets
        int row = i >> 6, c4 = i & 63;
        __builtin_amdgcn_global_load_async_to_lds_b128(
            (v4i*)(gsrc + (size_t)row * D_DIM + c4 * 4),
            (__attribute__((address_space(3))) v4i*)(lbase + row * HYS + c4 * 4),
            0, 0);
    }
#else
    for (int i = tid; i < CHUNK * 64; i += 256) {
        int row = i >> 6, c4 = i & 63;
        *(v4f*)(lbase + row * HYS + c4 * 4) =
            *(const v4f*)(gsrc + (size_t)row * D_DIM + c4 * 4);
    }
#endif
}

__global__ __launch_bounds__(256) void attn_ctx_kernel(
    const float* __restrict__ Hx,          // (B*LX) x 256, bias-free half
    const float* __restrict__ Hy,          // (B*LY) x 256, b1 folded in
    const float* __restrict__ Y,           // (B*LY) x 256 raw y
    const unsigned char* __restrict__ mask,// B x LY (bool)
    const float* __restrict__ W2,          // 256
    float* __restrict__ out)               // (B*LX) x 256
{
    extern __shared__ float smem[];
    float* sHx  = smem;                    // 16  * 260
    float* sHy0 = sHx  + 16 * HXS;         // 64  * 260  (double buffer A)
    float* sHy1 = sHy0 + CHUNK * HYS;      // 64  * 260  (double buffer B)
    float* sS   = sHy1 + CHUNK * HYS;      // 16  * 516  (scores -> probs)
    float* sW2  = sS + 16 * SSS;           // 256
    float* bufs[2] = { sHy0, sHy1 };

    const int tid  = threadIdx.x;
    const int lane = tid & 31;
    const int wv   = tid >> 5;

    const int b  = blockIdx.x / (LX_DIM / 16);
    const int xt = blockIdx.x % (LX_DIM / 16);

    const float* HxT = Hx + (size_t)(b * LX_DIM + xt * 16) * D_DIM;
    const float* HyB = Hy + (size_t)b * LY_DIM * D_DIM;
    const float* YB  = Y  + (size_t)b * LY_DIM * D_DIM;
    const unsigned char* mB = mask + b * LY_DIM;

    // ---- kick off Hy chunk 0 prefetch, then stage Hx tile (16x256) + W2 ----
    stage_chunk(HyB, bufs[0], tid);
    for (int i = tid; i < 16 * 64; i += 256) {          // 1024 v4f
        int row = i >> 6, c4 = i & 63;
        *(v4f*)(sHx + row * HXS + c4 * 4) =
            *(const v4f*)(HxT + row * D_DIM + c4 * 4);
    }
    if (tid < 64)
        *(v4f*)(sW2 + tid * 4) = *(const v4f*)(W2 + tid * 4);

    // ---- phase 1: scores tile (16 x 512) via VALU, masked, double-buffered ----
    const int yy  = tid & (CHUNK - 1);                  // chunk-local y
    const int xi0 = (tid >> 6) << 2;                    // 4 x-rows per thread
    const float NEG_INF = -__builtin_inff();
    const int NCH = LY_DIM / CHUNK;                     // 8

    for (int c = 0; c < NCH; ++c) {
        float* cur = bufs[c & 1];
        if (c + 1 < NCH) {                              // prefetch next chunk
            stage_chunk(HyB + (size_t)(c + 1) * CHUNK * D_DIM, bufs[(c + 1) & 1], tid);
            WAIT_ASYNC(16);                             // cur's 16 done (in-order)
        } else {
            WAIT_ASYNC(0);
        }
        __syncthreads();

        float a0 = 0.f, a1 = 0.f, a2 = 0.f, a3 = 0.f;
        for (int d = 0; d < D_DIM; d += 4) {
            v4f hy = *(const v4f*)(cur + yy * HYS + d);
            v4f w  = *(const v4f*)(sW2 + d);
            a0 += relu_dot4(*(const v4f*)(sHx + (xi0 + 0) * HXS + d), hy, w);
            a1 += relu_dot4(*(const v4f*)(sHx + (xi0 + 1) * HXS + d), hy, w);
            a2 += relu_dot4(*(const v4f*)(sHx + (xi0 + 2) * HXS + d), hy, w);
            a3 += relu_dot4(*(const v4f*)(sHx + (xi0 + 3) * HXS + d), hy, w);
        }
        const bool mk = mB[c * CHUNK + yy] != 0;        // mask==true -> -inf
        const int yg = c * CHUNK + yy;
        sS[(xi0 + 0) * SSS + yg] = mk ? NEG_INF : a0;   // b2 omitted: softmax-invariant
        sS[(xi0 + 1) * SSS + yg] = mk ? NEG_INF : a1;
        sS[(xi0 + 2) * SSS + yg] = mk ? NEG_INF : a2;
        sS[(xi0 + 3) * SSS + yg] = mk ? NEG_INF : a3;
        __syncthreads();                                // before cur is re-staged
    }

    // Prefetch Y chunk 0 now: latency hides under the softmax.
    stage_chunk(YB, bufs[0], tid);

    // ---- phase 2: wave32 softmax, 2 rows per wave ----
    for (int r = wv * 2; r < wv * 2 + 2; ++r) {
        float* srow = sS + r * SSS;
        float mx = NEG_INF;
        for (int yq = lane; yq < LY_DIM; yq += 32) mx = fmaxf(mx, srow[yq]);
        #pragma unroll
        for (int o = 16; o > 0; o >>= 1) mx = fmaxf(mx, __shfl_xor(mx, o));
        float sum = 0.f;
        for (int yq = lane; yq < LY_DIM; yq += 32) {
            float e = __expf(srow[yq] - mx);            // masked -> exp(-inf)=0
            srow[yq] = e;
            sum += e;
        }
        #pragma unroll
        for (int o = 16; o > 0; o >>= 1) sum += __shfl_xor(sum, o);
        const float inv = 1.0f / sum;
        for (int yq = lane; yq < LY_DIM; yq += 32) srow[yq] *= inv;
    }

    // ---- phase 3: contexts = P(16x512) @ Y_b(512x256), WMMA f32 16x16x4 ----
    const int mn = lane & 15;
    const int bk = (lane >> 4) << 1;
    const int nt0 = wv, nt1 = wv + 8;                   // 8 waves x 2 N-tiles
    v8f c0 = {}, c1 = {};

    for (int c = 0; c < NCH; ++c) {
        float* cur = bufs[c & 1];
        if (c + 1 < NCH) {
            stage_chunk(YB + (size_t)(c + 1) * CHUNK * D_DIM, bufs[(c + 1) & 1], tid);
            WAIT_ASYNC(16);
        } else {
            WAIT_ASYNC(0);
        }
        __syncthreads();                                // also fences softmax->reads

        for (int kk = 0; kk < CHUNK; kk += 4) {
            v2f a;
            a.x = sS[mn * SSS + c * CHUNK + kk + bk];
            a.y = sS[mn * SSS + c * CHUNK + kk + bk + 1];
            const float* br0 = cur + (kk + bk)     * HYS;
            const float* br1 = cur + (kk + bk + 1) * HYS;
            v2f b0; b0.x = br0[nt0 * 16 + mn]; b0.y = br1[nt0 * 16 + mn];
            c0 = __builtin_amdgcn_wmma_f32_16x16x4_f32(
                    false, a, false, b0, (short)0, c0, false, false);
            v2f b1; b1.x = br0[nt1 * 16 + mn]; b1.y = br1[nt1 * 16 + mn];
            c1 = __builtin_amdgcn_wmma_f32_16x16x4_f32(
                    false, a, false, b1, (short)0, c1, false, false);
        }
        __syncthreads();                                // before cur is re-staged
    }

    float* outT = out + (size_t)(b * LX_DIM + xt * 16) * D_DIM;
    const int mbase = (lane >> 4) << 3;
    #pragma unroll
    for (int r = 0; r < 8; ++r) {
        outT[(size_t)(mbase + r) * D_DIM + nt0 * 16 + mn] = c0[r];
        outT[(size_t)(mbase + r) * D_DIM + nt1 * 16 + mn] = c1[r];
    }
}

// ---------------------------------------------------------------------------
extern "C" void kernel_launch(void* const* d_in, const int* in_sizes, int n_in,
                              void* d_out, int out_size, void* d_ws, size_t ws_size,
                              hipStream_t stream) {
    (void)in_sizes; (void)n_in; (void)out_size; (void)ws_size;
    const float*         x      = (const float*)d_in[0];          // (8,128,256)
    const float*         y      = (const float*)d_in[1];          // (8,512,256)
    const unsigned char* y_mask = (const unsigned char*)d_in[2];  // (8,512) bool
    const float*         W1     = (const float*)d_in[3];          // (512,256)
    const float*         b1     = (const float*)d_in[4];          // (256)
    const float*         W2     = (const float*)d_in[5];          // (256,1)
    // d_in[6] = b2: constant shift of pre-softmax scores -> mathematically dead.
    float* out = (float*)d_out;

    float* Hx = (float*)d_ws;                                     // 1024 x 256
    float* Hy = Hx + (size_t)B_DIM * LX_DIM * D_DIM;              // 4096 x 256

    // Hx = x @ W1[:D]
    {
        const int Mtiles = (B_DIM * LX_DIM) / 16;                 // 64
        const int tiles  = Mtiles * (D_DIM / 16);                 // 1024
        h_gemm_kernel<<<tiles / 4, 128, 0, stream>>>(x, W1, nullptr, Hx, Mtiles);
    }
    // Hy = y @ W1[D:] + b1
    {
        const int Mtiles = (B_DIM * LY_DIM) / 16;                 // 256
        const int tiles  = Mtiles * (D_DIM / 16);                 // 4096
        h_gemm_kernel<<<tiles / 4, 128, 0, stream>>>(y, W1 + D_DIM * D_DIM, b1, Hy, Mtiles);
    }
    // fused scores -> softmax -> contexts (double-buffered async LDS staging)
    {
        const size_t smem = (size_t)(16 * HXS + 2 * CHUNK * HYS + 16 * SSS + 256) * sizeof(float);
        attn_ctx_kernel<<<B_DIM * (LX_DIM / 16), 256, smem, stream>>>(
            Hx, Hy, y, y_mask, W2, out);
    }
}